// E3TransferableAtomicOrbitals_42374147342463
// MI455X (gfx1250) — compile-verified
//
#include <hip/hip_runtime.h>
#include <hip/hip_bf16.h>

typedef __attribute__((ext_vector_type(2))) float v2f;
typedef __attribute__((ext_vector_type(8))) float v8f;

// D = A(16x4) * B(4x16) + C  -- V_WMMA_F32_16X16X4_F32
__device__ inline v8f wmma4(v2f a, v2f b, v8f c) {
    return __builtin_amdgcn_wmma_f32_16x16x4_f32(
        /*neg_a=*/false, a, /*neg_b=*/false, b,
        /*c_mod=*/(short)0, c, /*reuse_a=*/false, /*reuse_b=*/false);
}

// Generic strided f32 GEMM: C[m,n] = scale * sum_k A[m*lda + k*ACOL] * B[k*ldb + n] + bias[n]
// One 16x16 tile per wave. M%16==0, Ncols%16==0, K%4==0 required.
// ACOL is compile-time so the contiguous case vectorizes A loads to b64.
template <int ACOL>
__global__ void gemm_tile_f32(const float* __restrict__ A, int lda,
                              const float* __restrict__ B, int ldb,
                              float* __restrict__ C, int ldc,
                              const float* __restrict__ bias,
                              float scale, int M, int Ncols, int K)
{
    int lane = threadIdx.x & 31;
    int wave = threadIdx.x >> 5;
    int tilesN = Ncols >> 4;
    int tilesTotal = (M >> 4) * tilesN;
    int tile = blockIdx.x * (blockDim.x >> 5) + wave;
    if (tile >= tilesTotal) return;                  // wave-uniform guard
    int mt = tile / tilesN, nt = tile % tilesN;
    int mr = lane & 15;                              // A row / B col / C col (mod 16)
    int kh = (lane >> 4) << 1;                       // K sub-offset: 0 or 2
    int gm = mt * 16 + mr;                           // global A row
    int gn = nt * 16 + mr;                           // global B/C col
    v8f acc = {};
    for (int k = 0; k < K; k += 4) {
        v2f a, b;
        const float* ap = A + (size_t)gm * lda + (size_t)(k + kh) * ACOL;
        if (ACOL == 1) {
            a = *(const v2f*)ap;                     // 8B-aligned: lda even, k+kh even
        } else {
            a.x = ap[0];  a.y = ap[ACOL];
        }
        const float* bp = B + (size_t)(k + kh) * ldb + gn;
        b.x = bp[0];  b.y = bp[ldb];
        acc = wmma4(a, b, acc);
    }
    float bb = bias ? bias[gn] : 0.0f;
    int mbase = mt * 16 + (lane >> 4) * 8;           // C: lanes 0-15 -> M=r, 16-31 -> M=r+8
    for (int r = 0; r < 8; ++r)
        C[(size_t)(mbase + r) * ldc + gn] = scale * acc[r] + bb;
}

// One workgroup (256 threads) per (b,i). Builds h1 (64x64) via factored edge features,
// then h2 = tanh(h1 @ w2 + b2), edge_w = h2 @ w3 + b3 (8 cols, padded to 16).
__global__ void edge_mlp_kernel(const float* __restrict__ eattr,   // (4096,64) includes b1
                                const float* __restrict__ asr,     // (2048,64)
                                const float* __restrict__ ass,     // (2048,64)
                                const float* __restrict__ avr,     // (3,2048,64) incl 1/sqrt3
                                const float* __restrict__ avs,     // (3,2048,64) incl 1/sqrt3
                                const float* __restrict__ edge_sh, // (64,64,4)
                                const float* __restrict__ w2, const float* __restrict__ b2,
                                const float* __restrict__ w3, const float* __restrict__ b3,
                                float* __restrict__ edge_w)        // (32,64,64,8)
{
    __shared__ float h1[64 * 64];
    __shared__ float h2[64 * 64];
    int bi = blockIdx.x;
    int b = bi >> 6, i = bi & 63;
    int tid = threadIdx.x;

    // Step 1: h1[j,c] = tanh(Eattr[i,j,c] + y0*(AsRec[b,i,c]+AsSnd[b,j,c]) + sum_x y1x*(AvRec+AvSnd))
    for (int e = tid; e < 4096; e += 256) {
        int j = e >> 6, c = e & 63;
        const float* sh = edge_sh + ((size_t)i * 64 + j) * 4;
        float y0 = sh[0];
        float acc = eattr[((size_t)i * 64 + j) * 64 + c]
                  + y0 * (asr[((size_t)(b * 64 + i)) * 64 + c] +
                          ass[((size_t)(b * 64 + j)) * 64 + c]);
        for (int x = 0; x < 3; ++x) {
            acc += sh[1 + x] * (avr[(size_t)x * 131072 + (size_t)(b * 64 + i) * 64 + c] +
                                avs[(size_t)x * 131072 + (size_t)(b * 64 + j) * 64 + c]);
        }
        h1[e] = tanhf(acc);
    }
    __syncthreads();

    int lane = tid & 31, wave = tid >> 5;
    int mr = lane & 15, kh = (lane >> 4) << 1;

    // Step 2: h2 = tanh(h1 @ w2 + b2)   (64x64x64), 16 tiles over 8 waves
    for (int t = wave; t < 16; t += 8) {
        int mt = t >> 2, nt = t & 3;
        v8f acc = {};
        for (int k = 0; k < 64; k += 4) {
            v2f a = *(const v2f*)&h1[(mt * 16 + mr) * 64 + k + kh];   // ds_load_b64
            v2f bv;
            bv.x = w2[(k + kh) * 64 + nt * 16 + mr];
            bv.y = w2[(k + kh + 1) * 64 + nt * 16 + mr];
            acc = wmma4(a, bv, acc);
        }
        float bb = b2[nt * 16 + mr];
        int mbase = mt * 16 + (lane >> 4) * 8;
        for (int r = 0; r < 8; ++r)
            h2[(mbase + r) * 64 + nt * 16 + mr] = tanhf(acc[r] + bb);
    }
    __syncthreads();

    // Step 3: edge_w = h2 @ w3 + b3  (64x8, pad N to 16 with zero B columns)
    if (wave < 4) {                                   // wave-uniform
        int mt = wave;
        int n = mr;
        v8f acc = {};
        for (int k = 0; k < 64; k += 4) {
            v2f a = *(const v2f*)&h2[(mt * 16 + mr) * 64 + k + kh];
            v2f bv;
            bv.x = (n < 8) ? w3[(k + kh) * 8 + n] : 0.0f;
            bv.y = (n < 8) ? w3[(k + kh + 1) * 8 + n] : 0.0f;
            acc = wmma4(a, bv, acc);
        }
        if (n < 8) {
            float bb = b3[n];
            int mbase = mt * 16 + (lane >> 4) * 8;
            for (int r = 0; r < 8; ++r) {
                int j = mbase + r;
                edge_w[(((size_t)(b * 64 + i)) * 64 + j) * 8 + n] = acc[r] + bb;
            }
        }
    }
}

// Gather a per-head feature: cp<16 -> scalar part (p0), else vector part x=(cp-16)/16 (p1).
__device__ inline float gatherQKV(const float* __restrict__ p0, const float* __restrict__ p1,
                                  int row128, int h, int cp) {
    if (cp < 16) return p0[(size_t)row128 + h * 16 + cp];
    int x = (cp - 16) >> 4, ah = (cp - 16) & 15;
    return p1[(size_t)x * 262144 + (size_t)row128 + h * 16 + ah];
}

// One workgroup (256 threads) per (b,h): stage Q/K panels in LDS, logits = Q.K^T + edge_w,
// softmax over j (V staged into the Q buffer concurrently), U = attn.V, residual, scatter.
__global__ void attn_kernel(const float* __restrict__ q0, const float* __restrict__ q1,
                            const float* __restrict__ k0, const float* __restrict__ k1,
                            const float* __restrict__ v0, const float* __restrict__ v1,
                            const float* __restrict__ edge_w,
                            const float* __restrict__ node_s, const float* __restrict__ node_v,
                            float* __restrict__ out)
{
    __shared__ float Ls[64 * 64];
    __shared__ float Qs[64 * 64];   // reused for V in phase 3
    __shared__ float Ks[64 * 64];
    int bh = blockIdx.x;
    int b = bh >> 3, h = bh & 7;
    int tid = threadIdx.x, lane = tid & 31, wave = tid >> 5;
    int mr = lane & 15, kh = (lane >> 4) << 1;
    int bn = b * 64;

    // Stage Qh (scales pre-folded) and Kh panels: [row i/j][channel cp]
    for (int e = tid; e < 4096; e += 256) {
        int r = e >> 6, cp = e & 63;
        Qs[e] = gatherQKV(q0, q1, (bn + r) * 128, h, cp);
        Ks[e] = gatherQKV(k0, k1, (bn + r) * 128, h, cp);
    }
    __syncthreads();

    // Step 1: logits L = Q.K^T + edge_w
    for (int t = wave; t < 16; t += 8) {
        int mt = t >> 2, nt = t & 3;
        int jn = nt * 16 + mr;     // j for B lanes
        v8f acc = {};
        for (int k = 0; k < 64; k += 4) {
            v2f a  = *(const v2f*)&Qs[(mt * 16 + mr) * 64 + k + kh];
            v2f bv = *(const v2f*)&Ks[jn * 64 + k + kh];
            acc = wmma4(a, bv, acc);
        }
        int mbase = mt * 16 + (lane >> 4) * 8;
        for (int r = 0; r < 8; ++r) {
            int i = mbase + r;
            Ls[i * 64 + jn] = acc[r] + edge_w[(((size_t)(bn + i)) * 64 + jn) * 8 + h];
        }
    }
    __syncthreads();

    // Step 2: softmax over j per row i (threads 0-63) while threads 64+ help stage V into Qs.
    for (int e = tid; e < 4096; e += 256) {
        int r = e >> 6, cp = e & 63;
        Qs[e] = gatherQKV(v0, v1, (bn + r) * 128, h, cp);   // V panel overwrites Q panel
    }
    if (tid < 64) {
        float m = -1e30f;
        for (int j = 0; j < 64; ++j) m = fmaxf(m, Ls[tid * 64 + j]);
        float s = 0.0f;
        for (int j = 0; j < 64; ++j) { float e = __expf(Ls[tid * 64 + j] - m); Ls[tid * 64 + j] = e; s += e; }
        float inv = 1.0f / s;
        for (int j = 0; j < 64; ++j) Ls[tid * 64 + j] *= inv;
    }
    __syncthreads();

    // Step 3: U = attn @ V, residual, scatter
    for (int t = wave; t < 16; t += 8) {
        int mt = t >> 2, nt = t & 3;
        int cp = nt * 16 + mr;     // channel for B lanes
        v8f acc = {};
        for (int k = 0; k < 64; k += 4) {
            v2f a = *(const v2f*)&Ls[(mt * 16 + mr) * 64 + k + kh];
            v2f bv;
            bv.x = Qs[(k + kh) * 64 + cp];       // V[j=k+kh, cp]
            bv.y = Qs[(k + kh + 1) * 64 + cp];
            acc = wmma4(a, bv, acc);
        }
        int mbase = mt * 16 + (lane >> 4) * 8;
        for (int r = 0; r < 8; ++r) {
            int i = mbase + r;
            size_t orow = (size_t)(bn + i) * 512;
            if (cp < 16) {
                int ch = h * 16 + cp;
                out[orow + ch] = node_s[(size_t)(bn + i) * 128 + ch] + acc[r];
            } else {
                int x = (cp - 16) >> 4, ah = (cp - 16) & 15;
                int c = h * 16 + ah;
                out[orow + 128 + c * 3 + x] = node_v[((size_t)(bn + i) * 128 + c) * 3 + x] + acc[r];
            }
        }
    }
}

extern "C" void kernel_launch(void* const* d_in, const int* in_sizes, int n_in,
                              void* d_out, int out_size, void* d_ws, size_t ws_size,
                              hipStream_t stream) {
    const float* node_s    = (const float*)d_in[0];
    const float* node_v    = (const float*)d_in[1];
    const float* edge_attr = (const float*)d_in[2];
    const float* edge_sh   = (const float*)d_in[3];
    const float* wq0 = (const float*)d_in[4];
    const float* wq1 = (const float*)d_in[5];
    const float* wk0 = (const float*)d_in[6];
    const float* wk1 = (const float*)d_in[7];
    const float* wv0 = (const float*)d_in[8];
    const float* wv1 = (const float*)d_in[9];
    const float* w1  = (const float*)d_in[10];
    const float* b1  = (const float*)d_in[11];
    const float* w2  = (const float*)d_in[12];
    const float* b2  = (const float*)d_in[13];
    const float* w3  = (const float*)d_in[14];
    const float* b3  = (const float*)d_in[15];

    float* ws = (float*)d_ws;
    float* q0w   = ws;                 // 2048*128
    float* k0w   = ws + 262144;
    float* v0w   = ws + 524288;
    float* q1w   = ws + 786432;        // 3*2048*128
    float* k1w   = ws + 1572864;
    float* v1w   = ws + 2359296;
    float* asr   = ws + 3145728;       // 2048*64
    float* ass   = ws + 3276800;
    float* avr   = ws + 3407872;       // 3*2048*64
    float* avs   = ws + 3801088;
    float* eattr = ws + 4194304;       // 4096*64
    float* edgew = ws + 4456448;       // 32*64*64*8

    const float ATTN_NORM = 0.17677669529663687f;       // 1/sqrt(32)
    const float INV_SQRT3 = 0.57735026918962576f;
    const float QS  = ATTN_NORM;
    const float Q1S = ATTN_NORM * INV_SQRT3;

    dim3 blk(256);
    // node_s projections (M=2048, K=128, contiguous A)
    gemm_tile_f32<1><<<128, blk, 0, stream>>>(node_s, 128, wq0, 128, q0w, 128, nullptr, QS,   2048, 128, 128);
    gemm_tile_f32<1><<<128, blk, 0, stream>>>(node_s, 128, wk0, 128, k0w, 128, nullptr, 1.0f, 2048, 128, 128);
    gemm_tile_f32<1><<<128, blk, 0, stream>>>(node_s, 128, wv0, 128, v0w, 128, nullptr, 1.0f, 2048, 128, 128);
    gemm_tile_f32<1><<< 64, blk, 0, stream>>>(node_s, 128, w1 + 32  * 64, 64, asr, 64, nullptr, 1.0f, 2048, 64, 128);
    gemm_tile_f32<1><<< 64, blk, 0, stream>>>(node_s, 128, w1 + 288 * 64, 64, ass, 64, nullptr, 1.0f, 2048, 64, 128);
    // node_v component projections (element stride 3)
    for (int x = 0; x < 3; ++x) {
        const float* Av = node_v + x;
        gemm_tile_f32<3><<<128, blk, 0, stream>>>(Av, 384, wq1, 128, q1w + x * 262144, 128, nullptr, Q1S,  2048, 128, 128);
        gemm_tile_f32<3><<<128, blk, 0, stream>>>(Av, 384, wk1, 128, k1w + x * 262144, 128, nullptr, 1.0f, 2048, 128, 128);
        gemm_tile_f32<3><<<128, blk, 0, stream>>>(Av, 384, wv1, 128, v1w + x * 262144, 128, nullptr, 1.0f, 2048, 128, 128);
        gemm_tile_f32<3><<< 64, blk, 0, stream>>>(Av, 384, w1 + 160 * 64, 64, avr + x * 131072, 64, nullptr, INV_SQRT3, 2048, 64, 128);
        gemm_tile_f32<3><<< 64, blk, 0, stream>>>(Av, 384, w1 + 416 * 64, 64, avs + x * 131072, 64, nullptr, INV_SQRT3, 2048, 64, 128);
    }
    // Eattr = edge_attr @ w1[0:32] + b1   (M=4096, K=32, contiguous A)
    gemm_tile_f32<1><<<128, blk, 0, stream>>>(edge_attr, 32, w1, 64, eattr, 64, b1, 1.0f, 4096, 64, 32);

    // Edge MLP: one block per (b,i)
    edge_mlp_kernel<<<2048, blk, 0, stream>>>(eattr, asr, ass, avr, avs, edge_sh, w2, b2, w3, b3, edgew);

    // Attention: one block per (b,h)
    attn_kernel<<<256, blk, 0, stream>>>(q0w, q1w, k0w, k1w, v0w, v1w, edgew, node_s, node_v, (float*)d_out);
}